// GlobalContrastiveLoss_67886253081210
// MI455X (gfx1250) — compile-verified
//
#include <hip/hip_runtime.h>
#include <math.h>

typedef float v2f __attribute__((ext_vector_type(2)));
typedef float v8f __attribute__((ext_vector_type(8)));

#define BSZ 131072
#define DSZ 512
#define WAVES_PER_BLOCK 8
#define SAMPLES_PER_WAVE 16
#define SAMPLES_PER_BLOCK (WAVES_PER_BLOCK * SAMPLES_PER_WAVE)  // 128
#define NBLOCKS (BSZ / SAMPLES_PER_BLOCK)                       // 1024

// Main kernel: per-sample dot products via f32 WMMA (16x16x4), diagonal
// extraction, softplus loss, deterministic per-block reduction.
__global__ __launch_bounds__(256) void gcl_main_kernel(
    const float* __restrict__ img, const float* __restrict__ pos,
    const float* __restrict__ neg, float* __restrict__ partial) {
  const int tid  = threadIdx.x;
  const int lane = tid & 31;
  const int wave = tid >> 5;
  const int sampleBase = (blockIdx.x * WAVES_PER_BLOCK + wave) * SAMPLES_PER_WAVE;

  // WMMA f32 16x16x4 A-layout: lanes 0-15 -> K=0,1 ; lanes 16-31 -> K=2,3.
  // B (4x16) mirrors it, so the same per-lane float2 load feeds both A and B.
  const int row  = lane & 15;
  const int half = lane >> 4;  // 0 or 1
  const size_t rowBase = (size_t)(sampleBase + row) * DSZ + 2 * half;

  const float* ip = img + rowBase;
  const float* pp = pos + rowBase;
  const float* np = neg + rowBase;

  v8f cp = {0.f, 0.f, 0.f, 0.f, 0.f, 0.f, 0.f, 0.f};
  v8f cn = {0.f, 0.f, 0.f, 0.f, 0.f, 0.f, 0.f, 0.f};

#pragma unroll 8
  for (int k = 0; k < DSZ; k += 4) {
    v2f a  = *(const v2f*)(ip + k);
    v2f bp = *(const v2f*)(pp + k);
    v2f bn = *(const v2f*)(np + k);
    // D = A(16x4) x B(4x16) + C ; diagonal accumulates per-sample dots.
    cp = __builtin_amdgcn_wmma_f32_16x16x4_f32(
        /*neg_a=*/false, a, /*neg_b=*/false, bp, (short)0, cp, false, false);
    cn = __builtin_amdgcn_wmma_f32_16x16x4_f32(
        /*neg_a=*/false, a, /*neg_b=*/false, bn, (short)0, cn, false, false);
  }

  // C/D layout: VGPR r -> rows {r, r+8}; lanes 0-15 cols 0-15, lanes 16-31 cols 0-15.
  // Diagonal: sample m in 0..7  -> lane m,      element m
  //           sample m in 8..15 -> lane m+16,   element m-8
  union { v8f v; float f[8]; } up, un;
  up.v = cp;
  un.v = cn;

  float loss = 0.0f;
  if (lane < 8) {
    float dp = up.f[lane];
    float dn = un.f[lane];
    loss = log1pf(expf((dn - dp) * 0.02f));  // 1/50 temperature
  } else if (lane >= 24) {
    int i = lane - 24;
    float dp = up.f[i];
    float dn = un.f[i];
    loss = log1pf(expf((dn - dp) * 0.02f));
  }

  // Deterministic block tree-reduction (16 valid lanes per wave contribute).
  __shared__ float red[256];
  red[tid] = loss;
  __syncthreads();
#pragma unroll
  for (int s = 128; s > 0; s >>= 1) {
    if (tid < s) red[tid] += red[tid + s];
    __syncthreads();
  }
  if (tid == 0) partial[blockIdx.x] = red[0];
}

// Final deterministic reduction of 1024 block partials -> mean.
__global__ __launch_bounds__(256) void gcl_reduce_kernel(
    const float* __restrict__ partial, float* __restrict__ out) {
  __shared__ float red[256];
  float s = 0.0f;
#pragma unroll
  for (int i = threadIdx.x; i < NBLOCKS; i += 256) s += partial[i];
  red[threadIdx.x] = s;
  __syncthreads();
#pragma unroll
  for (int st = 128; st > 0; st >>= 1) {
    if (threadIdx.x < st) red[threadIdx.x] += red[threadIdx.x + st];
    __syncthreads();
  }
  if (threadIdx.x == 0) out[0] = red[0] * (1.0f / (float)BSZ);
}

extern "C" void kernel_launch(void* const* d_in, const int* in_sizes, int n_in,
                              void* d_out, int out_size, void* d_ws, size_t ws_size,
                              hipStream_t stream) {
  const float* img = (const float*)d_in[0];  // global_img [B, D, 1] f32
  const float* pos = (const float*)d_in[1];  // global_pos [B, D, 1] f32
  const float* neg = (const float*)d_in[2];  // global_neg [B, D, 1] f32
  float* out = (float*)d_out;                // scalar mean loss (f32)
  float* partial = (float*)d_ws;             // NBLOCKS floats of scratch

  gcl_main_kernel<<<NBLOCKS, 256, 0, stream>>>(img, pos, neg, partial);
  gcl_reduce_kernel<<<1, 256, 0, stream>>>(partial, out);
}